// ChamferBasedIndependent3dPoseAdv_56727928045822
// MI455X (gfx1250) — compile-verified
//
#include <hip/hip_runtime.h>
#include <math.h>

#define NF_CONST 4096
#define NR_CONST 8192
#define SEG 8            // column segments per row tile (occupancy multiplier)

typedef float v2f __attribute__((ext_vector_type(2)));
typedef float v8f __attribute__((ext_vector_type(8)));

__device__ __forceinline__ v8f wmma_k4(v2f a, v2f b, v8f c) {
  // D = A(16x4, f32) * B(4x16, f32) + C(16x16, f32)
  return __builtin_amdgcn_wmma_f32_16x16x4_f32(false, a, false, b, (short)0, c,
                                               false, false);
}

// Single-instruction min/max: skip clang's sNaN canonicalization (inputs here
// are finite — random normals / rotation matrices; f32 WMMA of finite data
// cannot produce NaN).
__device__ __forceinline__ float fmin_raw(float a, float b) {
  float d; asm("v_min_num_f32 %0, %1, %2" : "=v"(d) : "v"(a), "v"(b)); return d;
}
__device__ __forceinline__ float fmax_raw(float a, float b) {
  float d; asm("v_max_num_f32 %0, %1, %2" : "=v"(d) : "v"(a), "v"(b)); return d;
}

// Wave = (rowTile = blockIdx.x*4+wave, segment = blockIdx.y).
// Owns 16 rows of P, scans columns [seg*Nq/SEG, (seg+1)*Nq/SEG) of Q in
// 16-col tiles. Emits per-row partial min (squared translation dist) and
// partial max (trace(R_p R_q^T)) into out[seg*Np + row].
__global__ __launch_bounds__(128)
void pose_minmax_kernel(const float* __restrict__ Pt, const float* __restrict__ PR,
                        const float* __restrict__ Qt, const float* __restrict__ QR,
                        int Np, int Nq,
                        float* __restrict__ out_minsq, float* __restrict__ out_maxtr) {
  const int lane = threadIdx.x & 31;
  const int wave = threadIdx.x >> 5;
  const int m0   = (blockIdx.x * 4 + wave) * 16;
  const int seg  = blockIdx.y;
  const int l16  = lane & 15;
  const int half = lane >> 4;   // 0: K={0,1}, rows 0..7 ; 1: K={2,3}, rows 8..15
  const float keep = half ? 0.0f : 1.0f;   // zero-pad selector for K=8 chunk

  // ---------- loop-invariant A operands (row m = m0 + l16) ----------
  // Translation: A = (-2px, -2py, -2pz, 1); B = (qx, qy, qz, ||q||^2)
  //   => D = ||q||^2 - 2 p.q ; sq = D + ||p||^2 added once after the loop.
  const int m = m0 + l16;
  const float px = Pt[m * 3 + 0], py = Pt[m * 3 + 1], pz = Pt[m * 3 + 2];
  const float pn = px * px + py * py + pz * pz;
  v2f a_t;  a_t[0] = half ? -2.0f * pz : -2.0f * px;
            a_t[1] = half ?  1.0f      : -2.0f * py;
  const float* __restrict__ rp  = PR + (size_t)m * 9;
  const float* __restrict__ rpb = rp + 2 * half;           // address-select K half
  v2f a_r0; a_r0[0] = rpb[0]; a_r0[1] = rpb[1];            // K = 0..3
  v2f a_r1; a_r1[0] = rpb[4]; a_r1[1] = rpb[5];            // K = 4..7
  v2f a_r2; a_r2[0] = rp[8] * keep; a_r2[1] = 0.0f;        // K = 8, pad 9..11

  v8f minsq, maxtr;
#pragma unroll
  for (int i = 0; i < 8; ++i) { minsq[i] = 3.0e38f; maxtr[i] = -3.0e38f; }

  const int tilesPerSeg = Nq / (16 * SEG);
  const int t0 = seg * tilesPerSeg;
  for (int t = t0; t < t0 + tilesPerSeg; ++t) {
    const int n = t * 16 + l16;                            // column owned by lane
    const float qx = Qt[n * 3 + 0], qy = Qt[n * 3 + 1], qz = Qt[n * 3 + 2];
    const float qn = qx * qx + qy * qy + qz * qz;
    v2f b_t;  b_t[0] = half ? qz : qx;
              b_t[1] = half ? qn : qy;
    const float* __restrict__ rq  = QR + (size_t)n * 9;
    const float* __restrict__ rqb = rq + 2 * half;
    v2f b_r0; b_r0[0] = rqb[0]; b_r0[1] = rqb[1];
    v2f b_r1; b_r1[0] = rqb[4]; b_r1[1] = rqb[5];
    v2f b_r2; b_r2[0] = rq[8] * keep; b_r2[1] = 0.0f;

    // D[m,n] = ||q||^2 - 2 p.q   (C = inline 0)
    v8f c_t = {};
    c_t = wmma_k4(a_t, b_t, c_t);

    // tr[m,n] = sum_{k<9} RP[m,k] * RQ[n,k] via 3 chained K=4 WMMAs
    v8f c_r = {};
    c_r = wmma_k4(a_r0, b_r0, c_r);
    c_r = wmma_k4(a_r1, b_r1, c_r);
    c_r = wmma_k4(a_r2, b_r2, c_r);

#pragma unroll
    for (int i = 0; i < 8; ++i) {
      minsq[i] = fmin_raw(minsq[i], c_t[i]);
      maxtr[i] = fmax_raw(maxtr[i], c_r[i]);
    }
  }

  // Cross-lane reduce within each 16-lane half (rows 0..7 in lanes 0-15,
  // rows 8..15 in lanes 16-31; xor masks 1..8 never cross the halves).
#pragma unroll
  for (int mask = 1; mask <= 8; mask <<= 1) {
#pragma unroll
    for (int i = 0; i < 8; ++i) {
      minsq[i] = fmin_raw(minsq[i], __shfl_xor(minsq[i], mask, 32));
      maxtr[i] = fmax_raw(maxtr[i], __shfl_xor(maxtr[i], mask, 32));
    }
  }
  // Re-add the hoisted ||p||^2: accumulator element i (this lane-half) is row
  // m0 + 8*half + i, whose pn lives in lane 8*half + i.
#pragma unroll
  for (int i = 0; i < 8; ++i) {
    minsq[i] += __shfl(pn, (half << 3) + i, 32);
  }
  if (l16 == 0) {
    float* __restrict__ om = out_minsq + (size_t)seg * Np;
    float* __restrict__ ox = out_maxtr + (size_t)seg * Np;
#pragma unroll
    for (int i = 0; i < 8; ++i) {
      om[m0 + half * 8 + i] = minsq[i];
      ox[m0 + half * 8 + i] = maxtr[i];
    }
  }
}

__global__ __launch_bounds__(256)
void pose_final_reduce(const float* __restrict__ pm_R, const float* __restrict__ px_R,
                       const float* __restrict__ pm_F, const float* __restrict__ px_F,
                       float* __restrict__ out) {
  __shared__ float sdata[256];
  const float EPS_CH = 1e-5f, EPS_AC = 1e-6f;
  const float PI = 3.14159265358979323846f;
  float acc = 0.0f;
  for (int i = threadIdx.x; i < NR_CONST; i += 256) {
    float mn = pm_R[i], mx = px_R[i];
#pragma unroll
    for (int s = 1; s < SEG; ++s) {
      mn = fmin_raw(mn, pm_R[s * NR_CONST + i]);
      mx = fmax_raw(mx, px_R[s * NR_CONST + i]);
    }
    acc += sqrtf(fmaxf(mn, 0.0f) + EPS_CH) * (PI / (float)NR_CONST);
    float c = fminf(fmaxf(0.5f * (mx - 1.0f), -1.0f + EPS_AC), 1.0f - EPS_AC);
    acc += acosf(c) * (1.0f / (float)NR_CONST);
  }
  for (int i = threadIdx.x; i < NF_CONST; i += 256) {
    float mn = pm_F[i], mx = px_F[i];
#pragma unroll
    for (int s = 1; s < SEG; ++s) {
      mn = fmin_raw(mn, pm_F[s * NF_CONST + i]);
      mx = fmax_raw(mx, px_F[s * NF_CONST + i]);
    }
    acc += sqrtf(fmaxf(mn, 0.0f) + EPS_CH) * (PI / (float)NF_CONST);
    float c = fminf(fmaxf(0.5f * (mx - 1.0f), -1.0f + EPS_AC), 1.0f - EPS_AC);
    acc += acosf(c) * (1.0f / (float)NF_CONST);
  }
  sdata[threadIdx.x] = acc;
  __syncthreads();
  for (int s = 128; s > 0; s >>= 1) {
    if (threadIdx.x < s) sdata[threadIdx.x] += sdata[threadIdx.x + s];
    __syncthreads();
  }
  if (threadIdx.x == 0) out[0] = sdata[0];
}

extern "C" void kernel_launch(void* const* d_in, const int* in_sizes, int n_in,
                              void* d_out, int out_size, void* d_ws, size_t ws_size,
                              hipStream_t stream) {
  const float* t_fake = (const float*)d_in[0];   // [NF,3]
  const float* R_fake = (const float*)d_in[1];   // [NF,3,3]
  const float* r_buf  = (const float*)d_in[2];   // [NR,3,3]
  const float* t_buf  = (const float*)d_in[3];   // [1,NR,3] == [NR,3] flat

  float* ws   = (float*)d_ws;
  float* pm_R = ws;                                   // [SEG, NR] partial min sq
  float* px_R = pm_R + (size_t)SEG * NR_CONST;        // [SEG, NR] partial max tr
  float* pm_F = px_R + (size_t)SEG * NR_CONST;        // [SEG, NF]
  float* px_F = pm_F + (size_t)SEG * NF_CONST;        // [SEG, NF]

  // Pass 1: P = buffer rows. 512 row tiles x 8 segs, 4 waves/block.
  dim3 g1(NR_CONST / 16 / 4, SEG);
  pose_minmax_kernel<<<g1, 128, 0, stream>>>(
      t_buf, r_buf, t_fake, R_fake, NR_CONST, NF_CONST, pm_R, px_R);
  // Pass 2: P = fake rows. 256 row tiles x 8 segs.
  dim3 g2(NF_CONST / 16 / 4, SEG);
  pose_minmax_kernel<<<g2, 128, 0, stream>>>(
      t_fake, R_fake, t_buf, r_buf, NF_CONST, NR_CONST, pm_F, px_F);
  // Final scalar: rloss + tloss * pi  (with SEG-way combine folded in)
  pose_final_reduce<<<1, 256, 0, stream>>>(pm_R, px_R, pm_F, px_F, (float*)d_out);
}